// EdgeToEdgeMessagePassing_54597624267061
// MI455X (gfx1250) — compile-verified
//
#include <hip/hip_runtime.h>
#include <hip/hip_bf16.h>

// ---------------------------------------------------------------------------
// EdgeToEdge message passing for MI455X (gfx1250, wave32).
// Both GEMMs use V_WMMA_F32_16X16X4_F32 (fp32 exact). Scatter phases use
// global_atomic_add_f32. Everything is memory-bound (~3-4 GB traffic @ 23.3
// TB/s), so WMMA trivially covers the ~30 GFLOP of matrix work.
// ---------------------------------------------------------------------------

typedef float v2f __attribute__((ext_vector_type(2)));
typedef float v8f __attribute__((ext_vector_type(8)));

#define ND 256   // node feature dim
#define ED 128   // edge feature dim
#define BN_EPS 1e-5f

// ---------------------------------------------------------------------------
// Kernel 1: px[N,128] = x[N,256] @ W_proj[128,256]^T  (fp32 WMMA 16x16x4)
// Block = 256 threads = 8 waves; wave w owns col tile w*16; block owns 16 rows.
// A-frag (16x4 f32): lane L: row = L&15, K = k0 + 2*(L>>4) + {0,1}
// B-frag (4x16 f32): lane L: col = L&15, K = k0 + 2*(L>>4) + {0,1}; B[k][n]=W[n][k]
// D (16x16 f32):     vgpr r: M = r + 8*(L>>4), N = L&15
// ---------------------------------------------------------------------------
__global__ __launch_bounds__(256) void gemm_px_kernel(
    const float* __restrict__ X, const float* __restrict__ Wp,
    float* __restrict__ px, int Nrows)
{
    const int wave = threadIdx.x >> 5;
    const int lane = threadIdx.x & 31;
    const int r0   = blockIdx.x * 16;
    const int c0   = wave * 16;
    int arow = r0 + (lane & 15);
    if (arow >= Nrows) arow = Nrows - 1;            // clamp (tail only)
    const int kup = (lane >> 4) << 1;               // 0 or 2

    const float* __restrict__ Arow = X  + (size_t)arow * ND;
    const float* __restrict__ Brow = Wp + (size_t)(c0 + (lane & 15)) * ND;

    v8f acc = {};
    #pragma unroll 4
    for (int k0 = 0; k0 < ND; k0 += 4) {
        v2f a = *(const v2f*)(Arow + k0 + kup);     // global_load_b64
        v2f b = *(const v2f*)(Brow + k0 + kup);
        acc = __builtin_amdgcn_wmma_f32_16x16x4_f32(
            false, a, false, b, (short)0, acc, false, false);
    }

    const int m0 = (lane >> 4) << 3;
    const int nc = lane & 15;
    float* __restrict__ dbase = px + (size_t)(r0 + m0) * ED + c0 + nc;
    if (r0 + 16 <= Nrows) {                          // uniform: full strip
        #pragma unroll
        for (int r = 0; r < 8; ++r) dbase[(size_t)r * ED] = acc[r];
    } else {                                         // generic tail (unused here)
        for (int r = 0; r < 8; ++r)
            if (r0 + m0 + r < Nrows) dbase[(size_t)r * ED] = acc[r];
    }
}

// ---------------------------------------------------------------------------
// Kernel 2: fused[e,:] = edge_attr[e,:] + 0.5*(px[src[e],:] + px[dst[e],:])
// One wave (32 lanes * float4) per edge row; 8 rows per block.
// ---------------------------------------------------------------------------
__global__ __launch_bounds__(256) void fuse_kernel(
    const float* __restrict__ px, const float* __restrict__ ea,
    const int* __restrict__ ei, float* __restrict__ fused, int E)
{
    const int e    = blockIdx.x * 8 + (threadIdx.x >> 5);
    const int lane = threadIdx.x & 31;
    if (e >= E) return;
    const int src = ei[e];
    const int dst = ei[(size_t)E + e];
    float4 a = ((const float4*)(px + (size_t)src * ED))[lane];
    float4 b = ((const float4*)(px + (size_t)dst * ED))[lane];
    float4 c = ((const float4*)(ea + (size_t)e   * ED))[lane];
    float4 r;
    r.x = c.x + 0.5f * (a.x + b.x);
    r.y = c.y + 0.5f * (a.y + b.y);
    r.z = c.z + 0.5f * (a.z + b.z);
    r.w = c.w + 0.5f * (a.w + b.w);
    ((float4*)(fused + (size_t)e * ED))[lane] = r;
}

// ---------------------------------------------------------------------------
// Kernel 3: line-graph scatter-add: agg[line_dst] += fused[line_src]; cnt++
// One wave per line edge; 4 float atomics per lane (global_atomic_add_f32).
// ---------------------------------------------------------------------------
__global__ __launch_bounds__(256) void line_scatter_kernel(
    const float* __restrict__ fused, const int* __restrict__ lg,
    float* __restrict__ agg, float* __restrict__ cnt, int L)
{
    const int l    = blockIdx.x * 8 + (threadIdx.x >> 5);
    const int lane = threadIdx.x & 31;
    if (l >= L) return;
    const int s = lg[l];
    const int d = lg[(size_t)L + l];
    float4 v = ((const float4*)(fused + (size_t)s * ED))[lane];
    float* base = agg + (size_t)d * ED + lane * 4;
    atomicAdd(base + 0, v.x);
    atomicAdd(base + 1, v.y);
    atomicAdd(base + 2, v.z);
    atomicAdd(base + 3, v.w);
    if (lane == 0) atomicAdd(cnt + d, 1.0f);
}

// ---------------------------------------------------------------------------
// Kernel 4: h = PReLU((agg/max(cnt,1)) @ W1^T + b1), written IN PLACE over agg.
// Block owns a 16-row strip (exclusive); 8 waves = 8 col tiles. All A reads
// complete (data dep through wmma) before __syncthreads(), then stores.
// The 1/cnt segment-mean division is folded into the A-fragment load.
// ---------------------------------------------------------------------------
__global__ __launch_bounds__(256) void gemm_edge_kernel(
    float* __restrict__ AggH, const float* __restrict__ W1,
    const float* __restrict__ b1, const float* __restrict__ cnt,
    const float* __restrict__ prelu_a, int E)
{
    const int wave = threadIdx.x >> 5;
    const int lane = threadIdx.x & 31;
    const int r0   = blockIdx.x * 16;
    const int c0   = wave * 16;
    int arow = r0 + (lane & 15);
    if (arow >= E) arow = E - 1;
    const float invc = 1.0f / fmaxf(cnt[arow], 1.0f);
    const int kup = (lane >> 4) << 1;

    const float* __restrict__ Arow = AggH + (size_t)arow * ED;
    const float* __restrict__ Brow = W1   + (size_t)(c0 + (lane & 15)) * ED;

    v8f acc = {};
    #pragma unroll 4
    for (int k0 = 0; k0 < ED; k0 += 4) {
        v2f a = *(const v2f*)(Arow + k0 + kup);
        a.x *= invc; a.y *= invc;                   // fold segment-mean divide
        v2f b = *(const v2f*)(Brow + k0 + kup);
        acc = __builtin_amdgcn_wmma_f32_16x16x4_f32(
            false, a, false, b, (short)0, acc, false, false);
    }

    const float alpha = prelu_a[0];
    const int m0 = (lane >> 4) << 3;
    const int nc = lane & 15;
    const float bias = b1[c0 + nc];
    float res[8];
    #pragma unroll
    for (int r = 0; r < 8; ++r) {
        float v = acc[r] + bias;
        res[r] = (v >= 0.0f) ? v : alpha * v;       // PReLU
    }
    __syncthreads();                                 // all strip reads done
    float* __restrict__ dbase = AggH + (size_t)(r0 + m0) * ED + c0 + nc;
    if (r0 + 16 <= E) {                              // uniform: full strip
        #pragma unroll
        for (int r = 0; r < 8; ++r) dbase[(size_t)r * ED] = res[r];
    } else {
        for (int r = 0; r < 8; ++r)
            if (r0 + m0 + r < E) dbase[(size_t)r * ED] = res[r];
    }
}

// ---------------------------------------------------------------------------
// Kernel 5: per-channel BN statistics: sum(h), sum(h^2). Thread = channel,
// block-local accumulation then one atomicAdd pair per block per channel.
// ---------------------------------------------------------------------------
__global__ __launch_bounds__(128) void bn_stats_kernel(
    const float* __restrict__ h, float* __restrict__ bnsum,
    float* __restrict__ bnsumsq, int E)
{
    const int c = threadIdx.x;
    float s = 0.0f, ss = 0.0f;
    for (int e = blockIdx.x; e < E; e += gridDim.x) {
        float v = h[(size_t)e * ED + c];
        s += v; ss += v * v;
    }
    atomicAdd(&bnsum[c], s);
    atomicAdd(&bnsumsq[c], ss);
}

// Kernel 6: fold BN into scale/shift:  h_norm = h*scale + shift
__global__ __launch_bounds__(128) void bn_finalize_kernel(
    const float* __restrict__ bnsum, const float* __restrict__ bnsumsq,
    const float* __restrict__ gamma, const float* __restrict__ beta,
    float* __restrict__ scale, float* __restrict__ shift, int E)
{
    const int c = threadIdx.x;
    const float inv_n = 1.0f / (float)E;
    const float mean  = bnsum[c] * inv_n;
    float var = bnsumsq[c] * inv_n - mean * mean;   // biased, as torch BN
    var = fmaxf(var, 0.0f);
    const float sc = gamma[c] * rsqrtf(var + BN_EPS);
    scale[c] = sc;
    shift[c] = beta[c] - mean * sc;
}

// ---------------------------------------------------------------------------
// Kernel 7: out_row = fused + h*scale + shift, atomically scattered to nodes
// by dst; also counts edges per node. One wave per edge.
// ---------------------------------------------------------------------------
__global__ __launch_bounds__(256) void node_scatter_kernel(
    const float* __restrict__ fused, const float* __restrict__ h,
    const float* __restrict__ scale, const float* __restrict__ shift,
    const int* __restrict__ ei, float* __restrict__ node_acc,
    float* __restrict__ node_cnt, int E)
{
    const int e    = blockIdx.x * 8 + (threadIdx.x >> 5);
    const int lane = threadIdx.x & 31;
    if (e >= E) return;
    const int dst = ei[(size_t)E + e];
    float4 f  = ((const float4*)(fused + (size_t)e * ED))[lane];
    float4 hv = ((const float4*)(h     + (size_t)e * ED))[lane];
    float4 sc = ((const float4*)scale)[lane];
    float4 sh = ((const float4*)shift)[lane];
    float4 r;
    r.x = f.x + hv.x * sc.x + sh.x;
    r.y = f.y + hv.y * sc.y + sh.y;
    r.z = f.z + hv.z * sc.z + sh.z;
    r.w = f.w + hv.w * sc.w + sh.w;
    float* base = node_acc + (size_t)dst * ED + lane * 4;
    atomicAdd(base + 0, r.x);
    atomicAdd(base + 1, r.y);
    atomicAdd(base + 2, r.z);
    atomicAdd(base + 3, r.w);
    if (lane == 0) atomicAdd(node_cnt + dst, 1.0f);
}

// Kernel 8: node_updates /= max(count,1)   (in place on d_out)
__global__ __launch_bounds__(256) void node_mean_kernel(
    float* __restrict__ node_acc, const float* __restrict__ node_cnt, int total)
{
    const int i = blockIdx.x * 256 + threadIdx.x;
    if (i >= total) return;
    const int n = i >> 7;                            // / ED
    node_acc[i] *= 1.0f / fmaxf(node_cnt[n], 1.0f);
}

// ---------------------------------------------------------------------------
extern "C" void kernel_launch(void* const* d_in, const int* in_sizes, int n_in,
                              void* d_out, int out_size, void* d_ws, size_t ws_size,
                              hipStream_t stream)
{
    const float* x        = (const float*)d_in[0];
    const float* ea       = (const float*)d_in[1];
    const float* Wp       = (const float*)d_in[2];
    const float* W1       = (const float*)d_in[3];
    const float* b1       = (const float*)d_in[4];
    const float* prelu_a  = (const float*)d_in[5];
    const float* gamma    = (const float*)d_in[6];
    const float* beta     = (const float*)d_in[7];
    const int*   ei       = (const int*)d_in[8];
    const int*   lg       = (const int*)d_in[9];

    const int Nn = in_sizes[0] / ND;      // 50000
    const int E  = in_sizes[1] / ED;      // 800000
    const int L  = in_sizes[9] / 2;       // 1600000

    // ---- workspace carving (all offsets 16B aligned) ----
    char* w = (char*)d_ws;
    float* px    = (float*)w;  w += (size_t)Nn * ED * sizeof(float);
    float* fused = (float*)w;  w += (size_t)E  * ED * sizeof(float);
    float* aggh  = (float*)w;  w += (size_t)E  * ED * sizeof(float);  // agg, then h in-place
    float* cntL  = (float*)w;  w += (size_t)E  * sizeof(float);
    float* ncnt  = (float*)w;  w += (size_t)Nn * sizeof(float);
    float* bnsum = (float*)w;  w += ED * sizeof(float);
    float* bnssq = (float*)w;  w += ED * sizeof(float);
    float* bnsc  = (float*)w;  w += ED * sizeof(float);
    float* bnsh  = (float*)w;  w += ED * sizeof(float);

    float* node_acc = (float*)d_out;      // accumulate straight into output

    // ---- zero the accumulators (graph-capture-safe stream memsets) ----
    hipMemsetAsync(aggh,  0, (size_t)E * ED * sizeof(float), stream);
    hipMemsetAsync(cntL,  0, (size_t)E * sizeof(float), stream);
    hipMemsetAsync(ncnt,  0, (size_t)Nn * sizeof(float), stream);
    hipMemsetAsync(bnsum, 0, 2 * ED * sizeof(float), stream);   // sum + sumsq
    hipMemsetAsync(node_acc, 0, (size_t)Nn * ED * sizeof(float), stream);

    // 1) px = x @ W_proj^T   (WMMA f32)
    gemm_px_kernel<<<(Nn + 15) / 16, 256, 0, stream>>>(x, Wp, px, Nn);

    // 2) fused = edge_attr + 0.5*(px[src] + px[dst])
    fuse_kernel<<<(E + 7) / 8, 256, 0, stream>>>(px, ea, ei, fused, E);

    // 3) agg[line_dst] += fused[line_src]; count
    line_scatter_kernel<<<(L + 7) / 8, 256, 0, stream>>>(fused, lg, aggh, cntL, L);

    // 4) h = PReLU((agg/cnt) @ W1^T + b1), in place over agg  (WMMA f32)
    gemm_edge_kernel<<<(E + 15) / 16, 256, 0, stream>>>(aggh, W1, b1, cntL, prelu_a, E);

    // 5-6) batch-norm statistics, folded into scale/shift
    bn_stats_kernel<<<1024, 128, 0, stream>>>(aggh, bnsum, bnssq, E);
    bn_finalize_kernel<<<1, 128, 0, stream>>>(bnsum, bnssq, gamma, beta, bnsc, bnsh, E);

    // 7) scatter (fused + BN(h)) to nodes by dst
    node_scatter_kernel<<<(E + 7) / 8, 256, 0, stream>>>(
        fused, aggh, bnsc, bnsh, ei, node_acc, ncnt, E);

    // 8) divide by per-node counts
    node_mean_kernel<<<((size_t)Nn * ED + 255) / 256, 256, 0, stream>>>(
        node_acc, ncnt, Nn * ED);
}